// ThreeBodyInteration_56925496541390
// MI455X (gfx1250) — compile-verified
//
#include <hip/hip_runtime.h>
#include <hip/hip_bf16.h>
#include <math.h>

#define CUTOFF     5.0f
#define L_MAX      3
#define N_MAX      3
#define DEGREE     9
#define KPAD       12          // K padded to 3 k-steps of 4 for 16x16x4 WMMA
#define NUM_NODES  50000
#define NUM_EDGES  1000000
#define NUM_TRIPLETS 8000000
#define F_NODE     64
#define F_EDGE     64

typedef float v2f __attribute__((ext_vector_type(2)));
typedef float v8f __attribute__((ext_vector_type(8)));

// spherical Bessel zeros z_{l,n}
__constant__ float SBZ[3][3] = {
    {3.141592653589793f, 6.283185307179586f, 9.42477796076938f},
    {4.4934094579080615f, 7.725251836938652f, 10.904121659429897f},
    {5.763459196895549f, 9.09501133047736f, 12.322940970567323f},
};

__device__ __forceinline__ float fcut(float r) {
    float q = r * (1.0f / CUTOFF);
    float q2 = q * q;
    float q3 = q2 * q;
    float q4 = q2 * q2;
    float q5 = q4 * q;
    return 1.0f - 6.0f * q5 + 15.0f * q4 - 10.0f * q3;
}

__device__ __forceinline__ float sigmoidf_(float x) {
    return 1.0f / (1.0f + __expf(-x));
}

// ---------------------------------------------------------------- zero fill
__global__ void zero_kernel(float* __restrict__ p, long long n) {
    long long i = (long long)blockIdx.x * blockDim.x + threadIdx.x;
    if (i < n) p[i] = 0.0f;
}

// -------------------------- pack Wm/Wg into K-pair-interleaved padded layout
// Wpad layout: [pair p=0..5][n=0..63][2] ; pair p covers k = {2p, 2p+1},
// rows k >= 9 are zero. One v2f (b64) load fetches both K values for a lane.
__global__ void pack_weights_kernel(const float* __restrict__ Wm,
                                    const float* __restrict__ Wg,
                                    float* __restrict__ Wm_pad,
                                    float* __restrict__ Wg_pad) {
    int i = blockIdx.x * blockDim.x + threadIdx.x;   // over 6*64 = 384 pairs
    if (i >= 6 * F_EDGE) return;
    int p = i >> 6;
    int n = i & 63;
    int k0 = 2 * p, k1 = 2 * p + 1;
    float m0 = (k0 < DEGREE) ? Wm[k0 * F_EDGE + n] : 0.0f;
    float m1 = (k1 < DEGREE) ? Wm[k1 * F_EDGE + n] : 0.0f;
    float g0 = (k0 < DEGREE) ? Wg[k0 * F_EDGE + n] : 0.0f;
    float g1 = (k1 < DEGREE) ? Wg[k1 * F_EDGE + n] : 0.0f;
    Wm_pad[i * 2 + 0] = m0;
    Wm_pad[i * 2 + 1] = m1;
    Wg_pad[i * 2 + 0] = g0;
    Wg_pad[i * 2 + 1] = g1;
}

// ------------------------------------------------------ node gate (tiny GEMM)
__global__ __launch_bounds__(256)
void node_gate_kernel(const float* __restrict__ nf,
                      const float* __restrict__ W1,
                      const float* __restrict__ b1,
                      float* __restrict__ gate) {
    int node = blockIdx.x * blockDim.x + threadIdx.x;
    if (node >= NUM_NODES) return;
    float acc[DEGREE];
#pragma unroll
    for (int d = 0; d < DEGREE; ++d) acc[d] = b1[d];
    const float* row = nf + (size_t)node * F_NODE;
#pragma unroll 4
    for (int f = 0; f < F_NODE; ++f) {
        float v = row[f];
#pragma unroll
        for (int d = 0; d < DEGREE; ++d)
            acc[d] = fmaf(v, W1[f * DEGREE + d], acc[d]);
    }
#pragma unroll
    for (int d = 0; d < DEGREE; ++d)
        gate[(size_t)node * DEGREE + d] = sigmoidf_(acc[d]);
}

// --------------------------------------------- triplet stage: scatter-atomics
__global__ __launch_bounds__(256)
void triplet_kernel(const float* __restrict__ dist,
                    const float* __restrict__ angles,
                    const int* __restrict__ tei,     // [2, T] flat
                    const int* __restrict__ ei,      // [2, E] flat
                    const float* __restrict__ gate,  // [nodes, 9]
                    float* __restrict__ mid) {       // [edges, KPAD]
    long long t = (long long)blockIdx.x * blockDim.x + threadIdx.x;
    if (t >= NUM_TRIPLETS) return;

    int tij = tei[t];
    int tik = tei[(long long)NUM_TRIPLETS + t];
    float rij = dist[tij];
    float rik = dist[tik];
    float fc = fcut(rij) * fcut(rik);

    float x = angles[t];  // cos(theta)
    float P[3];
    P[0] = 1.0f;
    P[1] = x;
    P[2] = 0.5f * (3.0f * x * x - 1.0f);
    const float SPHC0 = 0.5641895835477563f;   // sqrt(1/pi)
    const float SPHC1 = 0.9772050238058398f;   // sqrt(3/pi)
    const float SPHC2 = 1.2615662610100802f;   // sqrt(5/pi)
    float sph[3] = {SPHC0 * P[0] * fc, SPHC1 * P[1] * fc, SPHC2 * P[2] * fc};

    int nodek = ei[NUM_EDGES + tik];           // edge_index[1][tik]
    const float* g = gate + (size_t)nodek * DEGREE;
    float* midrow = mid + (size_t)tij * KPAD;

    float scale = rik * (1.0f / CUTOFF);
#pragma unroll
    for (int l = 0; l < L_MAX; ++l) {
#pragma unroll
        for (int nn = 0; nn < N_MAX; ++nn) {
            float xx = SBZ[l][nn] * scale;     // xx >= ~0.31 always
            float s, c;
            __sincosf(xx, &s, &c);
            float inv = __frcp_rn(xx);
            float j0 = s * inv;
            float j;
            if (l == 0) {
                j = j0;
            } else {
                float j1 = (j0 - c) * inv;
                j = (l == 1) ? j1 : (3.0f * inv * j1 - j0);
            }
            float val = j * sph[l] * g[l * N_MAX + nn];
            atomicAdd(&midrow[l * N_MAX + nn], val);
        }
    }
}

// ---------------------- fused gated MLP + residual, f32 WMMA (16x16x4, K=12)
// All fragment loads are unconditional b64s (data pre-padded), so EXEC stays
// all-1s and no branchy predication surrounds the WMMA chain.
__global__ __launch_bounds__(128)
void edge_mlp_wmma_kernel(const float* __restrict__ mid,      // [E, KPAD]
                          const float* __restrict__ edge_attr,
                          const float* __restrict__ Wm_pad,   // [6][64][2]
                          const float* __restrict__ bm,
                          const float* __restrict__ Wg_pad,   // [6][64][2]
                          const float* __restrict__ bg,
                          float* __restrict__ out) {
    const int lane  = threadIdx.x & 31;
    const int wave  = threadIdx.x >> 5;
    const int tile  = blockIdx.x * 4 + wave;   // one 16-edge tile per wave
    const int edge0 = tile * 16;
    if (edge0 >= NUM_EDGES) return;            // wave-uniform

    const int mrow  = lane & 15;               // A: M = lane & 15
    const int khalf = (lane >> 4) << 1;        // K sub-offset: 0 or 2
    const int ncol  = lane & 15;               // B/D: N = lane & 15
    const int phase = lane >> 4;               // 0 or 1 (K-pair selector)

    // A fragments: padded mid tile [16 x 12], 3 k-steps of 4 -> 3 b64 loads
    const float* midrow = mid + (size_t)(edge0 + mrow) * KPAD;
    v2f a[3];
#pragma unroll
    for (int ks = 0; ks < 3; ++ks)
        a[ks] = *(const v2f*)(midrow + ks * 4 + khalf);

#pragma unroll
    for (int nt = 0; nt < 4; ++nt) {
        const int nbase = nt * 16 + ncol;
        v8f accm = {};
        v8f accg = {};
#pragma unroll
        for (int ks = 0; ks < 3; ++ks) {
            const int p = ks * 2 + phase;      // K-pair index 0..5
            const size_t off = ((size_t)p * F_EDGE + nbase) * 2;
            v2f bmf = *(const v2f*)(Wm_pad + off);
            v2f bgf = *(const v2f*)(Wg_pad + off);
            accm = __builtin_amdgcn_wmma_f32_16x16x4_f32(
                false, a[ks], false, bmf, (short)0, accm, false, false);
            accg = __builtin_amdgcn_wmma_f32_16x16x4_f32(
                false, a[ks], false, bgf, (short)0, accg, false, false);
        }
        const float biasm = bm[nbase];
        const float biasg = bg[nbase];
        const int mhi = phase << 3;            // 0 or 8
#pragma unroll
        for (int r = 0; r < 8; ++r) {
            int e = edge0 + r + mhi;           // D: M = r + 8*(lane>=16)
            float hm = accm[r] + biasm;
            float hg = accg[r] + biasg;
            float upd = hm * sigmoidf_(hm) * sigmoidf_(hg);  // silu * sigmoid
            size_t idx = (size_t)e * F_EDGE + nbase;
            out[idx] = edge_attr[idx] + upd;
        }
    }
}

extern "C" void kernel_launch(void* const* d_in, const int* in_sizes, int n_in,
                              void* d_out, int out_size, void* d_ws, size_t ws_size,
                              hipStream_t stream) {
    const float* edge_distances = (const float*)d_in[0];
    const float* triplet_angles = (const float*)d_in[1];
    const float* node_features  = (const float*)d_in[2];
    const float* edge_attr      = (const float*)d_in[3];
    const float* W1             = (const float*)d_in[4];
    const float* b1             = (const float*)d_in[5];
    const float* Wm             = (const float*)d_in[6];
    const float* bm             = (const float*)d_in[7];
    const float* Wg             = (const float*)d_in[8];
    const float* bg             = (const float*)d_in[9];
    const int*   tei            = (const int*)d_in[10];
    const int*   ei             = (const int*)d_in[11];
    float*       out            = (float*)d_out;

    // workspace layout: mid [E x KPAD] | gate [nodes x 9] | Wm_pad | Wg_pad
    float* mid    = (float*)d_ws;
    float* gate   = mid  + (size_t)NUM_EDGES * KPAD;
    float* Wm_pad = gate + (size_t)NUM_NODES * DEGREE;
    float* Wg_pad = Wm_pad + 6 * F_EDGE * 2;

    // 1) zero padded mid accumulator (every call; graph-replay safe)
    {
        long long n = (long long)NUM_EDGES * KPAD;
        int blocks = (int)((n + 255) / 256);
        zero_kernel<<<blocks, 256, 0, stream>>>(mid, n);
    }
    // 2) pack padded weights (tiny)
    {
        pack_weights_kernel<<<2, 256, 0, stream>>>(Wm, Wg, Wm_pad, Wg_pad);
    }
    // 3) node gate table
    {
        int blocks = (NUM_NODES + 255) / 256;
        node_gate_kernel<<<blocks, 256, 0, stream>>>(node_features, W1, b1, gate);
    }
    // 4) triplet scatter (L2-resident atomics)
    {
        int blocks = (NUM_TRIPLETS + 255) / 256;
        triplet_kernel<<<blocks, 256, 0, stream>>>(
            edge_distances, triplet_angles, tei, ei, gate, mid);
    }
    // 5) fused gated MLP + residual via f32 WMMA
    {
        int tiles = NUM_EDGES / 16;            // 62500 (divides exactly)
        int blocks = (tiles + 3) / 4;          // 4 waves per block
        edge_mlp_wmma_kernel<<<blocks, 128, 0, stream>>>(
            mid, edge_attr, Wm_pad, bm, Wg_pad, bg, out);
    }
}